// PermutationInvariantLoss_12584254177894
// MI455X (gfx1250) — compile-verified
//
#include <hip/hip_runtime.h>

typedef __attribute__((ext_vector_type(4))) float f4;
typedef __attribute__((ext_vector_type(2))) float v2f;
typedef __attribute__((ext_vector_type(8))) float v8f;

#define B_DIM 32
#define T_DIM 512
#define F_DIM 513
#define S_DIM 3

constexpr int    GROUPS_PER_BATCH = T_DIM * F_DIM;              // 262656 (b,n) groups per batch
constexpr int    QUADS_PER_BATCH  = GROUPS_PER_BATCH / 4;       // 65664 (12-float quads, exact)
constexpr size_t FLOATS_PER_BATCH = (size_t)GROUPS_PER_BATCH * S_DIM;

#define TPB 256
#define BLOCKS_PER_BATCH 64
#define NBLOCKS (B_DIM * BLOCKS_PER_BATCH)

// ws layout (floats):
//   [0, NBLOCKS*9)               per-block partial sums of |e_i - t_j|
//   [SUMS_OFFSET, +B_DIM*12)     per-batch sums, padded to K=12 (entries 9..11 stay 0)
constexpr int SUMS_OFFSET = NBLOCKS * 9;

// ---------------------------------------------------------------------------
// Kernel 1: HBM-bound streaming pass. Each thread consumes quads of 4 source
// groups (12 contiguous floats) from each tensor via b128 non-temporal loads
// and accumulates the 3x3 |est_i - tgt_j| sums in registers. Deterministic
// LDS tree reduction per block; one 9-float partial row per block.
// ---------------------------------------------------------------------------
__global__ void __launch_bounds__(TPB) pil_partial(const float* __restrict__ est,
                                                   const float* __restrict__ tgt,
                                                   float* __restrict__ ws) {
  const int b   = blockIdx.x / BLOCKS_PER_BATCH;
  const int blk = blockIdx.x % BLOCKS_PER_BATCH;
  const int tid = threadIdx.x;

  const f4* e4 = reinterpret_cast<const f4*>(est + (size_t)b * FLOATS_PER_BATCH);
  const f4* t4 = reinterpret_cast<const f4*>(tgt + (size_t)b * FLOATS_PER_BATCH);

  float acc[9];
#pragma unroll
  for (int k = 0; k < 9; ++k) acc[k] = 0.0f;

  for (int q = blk * TPB + tid; q < QUADS_PER_BATCH; q += BLOCKS_PER_BATCH * TPB) {
    const size_t base = 3 * (size_t)q;
    f4 ea = __builtin_nontemporal_load(e4 + base + 0);
    f4 eb = __builtin_nontemporal_load(e4 + base + 1);
    f4 ec = __builtin_nontemporal_load(e4 + base + 2);
    f4 ta = __builtin_nontemporal_load(t4 + base + 0);
    f4 tb = __builtin_nontemporal_load(t4 + base + 1);
    f4 tc = __builtin_nontemporal_load(t4 + base + 2);

    float e[12] = {ea.x, ea.y, ea.z, ea.w, eb.x, eb.y, eb.z, eb.w, ec.x, ec.y, ec.z, ec.w};
    float t[12] = {ta.x, ta.y, ta.z, ta.w, tb.x, tb.y, tb.z, tb.w, tc.x, tc.y, tc.z, tc.w};

#pragma unroll
    for (int g = 0; g < 4; ++g)
#pragma unroll
      for (int i = 0; i < 3; ++i)
#pragma unroll
        for (int j = 0; j < 3; ++j)
          acc[i * 3 + j] += __builtin_fabsf(e[g * 3 + i] - t[g * 3 + j]);
  }

  __shared__ float red[9][TPB];
#pragma unroll
  for (int k = 0; k < 9; ++k) red[k][tid] = acc[k];
  __syncthreads();

  for (int s = TPB / 2; s > 0; s >>= 1) {
    if (tid < s) {
#pragma unroll
      for (int k = 0; k < 9; ++k) red[k][tid] += red[k][tid + s];
    }
    __syncthreads();
  }
  if (tid < 9) ws[(size_t)blockIdx.x * 9 + tid] = red[tid][0];
}

// ---------------------------------------------------------------------------
// Kernel 2: deterministic reduction of the 64 per-block partials of each
// batch into a padded 32x12 sum table (K padded to 12 for the WMMA chunks).
// ---------------------------------------------------------------------------
__global__ void pil_reduce(float* __restrict__ ws) {
  const int i = blockIdx.x * blockDim.x + threadIdx.x;  // 0..383
  if (i >= B_DIM * 12) return;
  const int b = i / 12;
  const int k = i % 12;
  float s = 0.0f;
  if (k < 9) {
    const float* p = ws + (size_t)b * BLOCKS_PER_BATCH * 9 + k;
    for (int blk = 0; blk < BLOCKS_PER_BATCH; ++blk) s += p[(size_t)blk * 9];
  }
  ws[SUMS_OFFSET + i] = s;
}

// ---------------------------------------------------------------------------
// Kernel 3: single-wave finalize. losses(6x32) = P(6x9) x S(9x32) * 1/(3N)
// computed with chained V_WMMA_F32_16X16X4_F32 (K=12 in 3 chunks, two 16-col
// tiles). A layout (16x4 f32): VGPR0 holds K={0|2}, VGPR1 K={1|3} per lane
// half; B (4x16) rows half-split across lane groups; D rows m in d[m] on
// lanes 0-15 (rows >=6 are zero since P rows are zero there).
// Permutation masks: bit k=i*3+j set iff perm picks (i=p[j], j).
// ---------------------------------------------------------------------------
__global__ void __launch_bounds__(32) pil_final(const float* __restrict__ sums,
                                                float* __restrict__ out) {
  const int  lane = threadIdx.x;
  const int  M    = lane & 15;
  const bool hi   = lane >= 16;
  const float scale = 1.0f / (3.0f * (float)GROUPS_PER_BATCH);

  // perms of (0,1,2): {0,4,8},{0,5,7},{1,3,8},{2,3,7},{1,5,6},{2,4,6}
  const unsigned kPermMask[6] = {0x111u, 0x0A1u, 0x10Au, 0x08Cu, 0x062u, 0x054u};
  const unsigned msk = (M < 6) ? kPermMask[M] : 0u;

  v2f a[3];
#pragma unroll
  for (int c = 0; c < 3; ++c) {
    const int k0 = 4 * c + (hi ? 2 : 0);
    a[c].x = ((msk >> k0) & 1u) ? scale : 0.0f;
    a[c].y = ((msk >> (k0 + 1)) & 1u) ? scale : 0.0f;
  }

  float batchmin[2];
#pragma unroll
  for (int t = 0; t < 2; ++t) {
    const int batch = t * 16 + M;
    v8f d = {};
#pragma unroll
    for (int c = 0; c < 3; ++c) {
      const int r0 = 4 * c + (hi ? 2 : 0);
      v2f bb;
      bb.x = sums[batch * 12 + r0];
      bb.y = sums[batch * 12 + r0 + 1];
      d = __builtin_amdgcn_wmma_f32_16x16x4_f32(false, a[c], false, bb,
                                                (short)0, d, false, false);
    }
    float mn = d[0];
    mn = fminf(mn, d[1]);
    mn = fminf(mn, d[2]);
    mn = fminf(mn, d[3]);
    mn = fminf(mn, d[4]);
    mn = fminf(mn, d[5]);
    batchmin[t] = mn;
  }

  float tot = hi ? 0.0f : (batchmin[0] + batchmin[1]);
#pragma unroll
  for (int off = 16; off > 0; off >>= 1) tot += __shfl_down(tot, off, 32);
  if (lane == 0) out[0] = tot * (1.0f / (float)B_DIM);
}

extern "C" void kernel_launch(void* const* d_in, const int* in_sizes, int n_in,
                              void* d_out, int out_size, void* d_ws, size_t ws_size,
                              hipStream_t stream) {
  (void)in_sizes; (void)n_in; (void)out_size; (void)ws_size;
  const float* est = (const float*)d_in[0];
  const float* tgt = (const float*)d_in[1];
  float* ws  = (float*)d_ws;
  float* out = (float*)d_out;

  pil_partial<<<NBLOCKS, TPB, 0, stream>>>(est, tgt, ws);
  pil_reduce<<<2, 192, 0, stream>>>(ws);
  pil_final<<<1, 32, 0, stream>>>(ws + SUMS_OFFSET, out);
}